// NTXentLoss_11089605559125
// MI455X (gfx1250) — compile-verified
//
#include <hip/hip_runtime.h>
#include <math.h>

typedef __attribute__((ext_vector_type(2))) float v2f;
typedef __attribute__((ext_vector_type(8))) float v8f;
typedef __attribute__((ext_vector_type(4))) float f4;
typedef __attribute__((ext_vector_type(4))) unsigned int u32x4;
typedef __attribute__((ext_vector_type(8))) int i32x8;
typedef __attribute__((ext_vector_type(4))) int i32x4;

#define BHALF 4096          // B
#define N2    8192          // 2B rows
#define DDIM  256           // feature dim
#define TEMP_INV 10.0f      // 1 / 0.1

#define ROWS_PB  128                    // rows per block (A panel)
#define COLS_PI  64                     // cols staged per iteration (B panel)
#define CSPLIT   4                      // column-sweep split (gridDim.y)
#define NT_TOTAL (N2 / COLS_PI)         // 128 column tiles
#define NT_PER   (NT_TOTAL / CSPLIT)    // 32 per block

#if __has_builtin(__builtin_amdgcn_tensor_load_to_lds) && \
    __has_builtin(__builtin_amdgcn_s_wait_tensorcnt)
#define HAVE_TDM 1
#else
#define HAVE_TDM 0
#endif

#if HAVE_TDM
// DMA `nrows` consecutive 256-float rows (row-major, contiguous) from global
// into LDS via the Tensor Data Mover. 2D D# tile: 256 x nrows, 4-byte elems.
__device__ __forceinline__ void tdm_load_rows(const float* gsrc, void* ldst, int nrows) {
    unsigned long long ga = (unsigned long long)(uintptr_t)gsrc;
    unsigned lds = (unsigned)(uintptr_t)ldst;   // addr[31:0] == LDS byte offset

    u32x4 g0;
    g0[0] = 1u;                                              // count=1, user mode
    g0[1] = lds;                                             // lds_addr (bytes)
    g0[2] = (unsigned)(ga & 0xffffffffull);                  // global_addr[31:0]
    g0[3] = (unsigned)((ga >> 32) & 0x01ffffffull)           // global_addr[56:32]
          | (2u << 30);                                      // type = 2 ("image")

    i32x8 g1;
    g1[0] = 0x00020000;            // workgroup_mask=0, data_size=2 (4B)
    g1[1] = (int)(DDIM << 16);     // tensor_dim0[15:0] in bits 63:48 (=256)
    g1[2] = (int)(nrows << 16);    // tensor_dim0[31:16]=0 | tensor_dim1[15:0]
    g1[3] = (int)(DDIM << 16);     // tensor_dim1[31:16]=0 | tile_dim0 (=256)
    g1[4] = nrows;                 // tile_dim1 | tile_dim2=0
    g1[5] = DDIM;                  // tensor_dim0_stride[31:0] (=256 elems)
    g1[6] = 0;                     // stride0[47:32] | dim1_stride[15:0]
    g1[7] = 0;                     // dim1_stride[47:16]

    i32x4 z4 = {0, 0, 0, 0};
#if __clang_major__ >= 23
    i32x8 z8 = {0, 0, 0, 0, 0, 0, 0, 0};
    __builtin_amdgcn_tensor_load_to_lds(g0, g1, z4, z4, z8, 0);
#else
    __builtin_amdgcn_tensor_load_to_lds(g0, g1, z4, z4, 0);
#endif
}
#endif

// ---------------------------------------------------------------------------
// Kernel 1: row-normalize reps = cat([zjs, zis]) -> rn; also zero rowsum.
// ---------------------------------------------------------------------------
__global__ void __launch_bounds__(256)
k_normalize(const float* __restrict__ zis, const float* __restrict__ zjs,
            float* __restrict__ rn, float* __restrict__ rowsum) {
    const int wave = threadIdx.x >> 5;
    const int lane = threadIdx.x & 31;
    const int row  = blockIdx.x * 8 + wave;

    const int gid = blockIdx.x * 256 + threadIdx.x;
    if (gid < N2) rowsum[gid] = 0.0f;

    const float* src = (row < BHALF) ? (zjs + (size_t)row * DDIM)
                                     : (zis + (size_t)(row - BHALF) * DDIM);
    float v[8];
    float ss = 0.0f;
#pragma unroll
    for (int k = 0; k < 8; ++k) {
        v[k] = src[lane + 32 * k];
        ss += v[k] * v[k];
    }
#pragma unroll
    for (int m = 16; m >= 1; m >>= 1) ss += __shfl_xor(ss, m, 32);

    const float inv = 1.0f / fmaxf(sqrtf(ss), 1e-8f);
    float* dst = rn + (size_t)row * DDIM;
#pragma unroll
    for (int k = 0; k < 8; ++k) dst[lane + 32 * k] = v[k] * inv;
}

// ---------------------------------------------------------------------------
// Kernel 2: fused Gram sweep. Block = 8 waves (4 row-groups x 2 col-groups),
// 128 rows x 64-col iterations. Each wave computes a 32x32 tile with 4
// f32-WMMA accumulators. A panel (128KB) + double-buffered B panels (2x64KB)
// staged by the Tensor Data Mover; exp/rowsum/diag/pos fused per tile.
// ---------------------------------------------------------------------------
__global__ void __launch_bounds__(256)
k_gram(const float* __restrict__ rn,
       float* __restrict__ rowsum,
       float* __restrict__ diag,
       float* __restrict__ pos) {
    __shared__ float As[ROWS_PB * DDIM];       // 128 KB
    __shared__ float Bs[2][COLS_PI * DDIM];    // 2 x 64 KB
    __shared__ float blockRowSum[ROWS_PB];

    const int tid     = threadIdx.x;
    const int wave    = tid >> 5;
    const int lane    = tid & 31;
    const int wr      = wave >> 1;             // 0..3 : 32-row group
    const int wc      = wave & 1;              // 0..1 : 32-col group
    const int rowBase = blockIdx.x * ROWS_PB;
    const int ctBase  = blockIdx.y * NT_PER;

    if (tid < ROWS_PB) blockRowSum[tid] = 0.0f;

#if HAVE_TDM
    // wave-uniform predicate -> scalar branch; TDM ignores EXEC, so gate by wave
    const bool waveZero = (__builtin_amdgcn_readfirstlane((int)tid) >> 5) == 0;
    if (waveZero) {
        tdm_load_rows(rn + (size_t)rowBase * DDIM, As, ROWS_PB);
        tdm_load_rows(rn + (size_t)ctBase * COLS_PI * DDIM, &Bs[0][0], COLS_PI);
    }
    __builtin_amdgcn_s_wait_tensorcnt(0);
#else
    {   // cooperative A stage: 8192 float4, 32 per thread
        const f4* src = (const f4*)(rn + (size_t)rowBase * DDIM);
        f4* dst = (f4*)As;
#pragma unroll
        for (int i = 0; i < 32; ++i) dst[tid + 256 * i] = src[tid + 256 * i];
    }
#endif
    __syncthreads();

    const int half = lane >> 4;        // k-pair select within the 4-wide K step
    const int lid  = lane & 15;        // row-in-tile (A) / col-in-tile (B)
    const int kOff = 2 * half;

    const float* A0 = As + (wr * 32 + lid) * DDIM + kOff;   // tile rows 0-15
    const float* A1 = A0 + 16 * DDIM;                       // tile rows 16-31
    const int iBase = rowBase + wr * 32;

    float acc0[8], acc1[8];
#pragma unroll
    for (int r = 0; r < 8; ++r) { acc0[r] = 0.0f; acc1[r] = 0.0f; }

    for (int t = 0; t < NT_PER; ++t) {
        const int cur = t & 1;
#if HAVE_TDM
        if (t + 1 < NT_PER && waveZero)     // prefetch next panel during compute
            tdm_load_rows(rn + (size_t)(ctBase + t + 1) * COLS_PI * DDIM,
                          &Bs[cur ^ 1][0], COLS_PI);
#else
        {   // cooperative B stage: 4096 float4, 16 per thread
            const f4* src = (const f4*)(rn + (size_t)(ctBase + t) * COLS_PI * DDIM);
            f4* dst = (f4*)&Bs[cur][0];
#pragma unroll
            for (int i = 0; i < 16; ++i) dst[tid + 256 * i] = src[tid + 256 * i];
        }
        __syncthreads();
#endif
        const float* B0 = &Bs[cur][(wc * 32 + lid) * DDIM + kOff];  // cols 0-15
        const float* B1 = B0 + 16 * DDIM;                           // cols 16-31

        v8f c00 = {0.f,0.f,0.f,0.f,0.f,0.f,0.f,0.f};
        v8f c01 = {0.f,0.f,0.f,0.f,0.f,0.f,0.f,0.f};
        v8f c10 = {0.f,0.f,0.f,0.f,0.f,0.f,0.f,0.f};
        v8f c11 = {0.f,0.f,0.f,0.f,0.f,0.f,0.f,0.f};
#pragma unroll
        for (int k0 = 0; k0 < DDIM; k0 += 4) {
            v2f a0 = *(const v2f*)(A0 + k0);
            v2f a1 = *(const v2f*)(A1 + k0);
            v2f b0 = *(const v2f*)(B0 + k0);
            v2f b1 = *(const v2f*)(B1 + k0);
            c00 = __builtin_amdgcn_wmma_f32_16x16x4_f32(false, a0, false, b0, (short)0, c00, false, false);
            c01 = __builtin_amdgcn_wmma_f32_16x16x4_f32(false, a0, false, b1, (short)0, c01, false, false);
            c10 = __builtin_amdgcn_wmma_f32_16x16x4_f32(false, a1, false, b0, (short)0, c10, false, false);
            c11 = __builtin_amdgcn_wmma_f32_16x16x4_f32(false, a1, false, b1, (short)0, c11, false, false);
        }

        // C/D layout: VGPR r -> (M = r + 8*half, N = lid) within each 16x16.
        const int jB = (ctBase + t) * COLS_PI + wc * 32;
        const int j0 = jB + lid;
        const int j1 = jB + 16 + lid;
#pragma unroll
        for (int r = 0; r < 8; ++r) {
            const int i0 = iBase + r + 8 * half;
            const int i1 = i0 + 16;
            const int p0 = (i0 + BHALF) & (N2 - 1);
            const int p1 = (i1 + BHALF) & (N2 - 1);
            const float e00 = __expf(c00[r] * TEMP_INV);
            const float e01 = __expf(c01[r] * TEMP_INV);
            const float e10 = __expf(c10[r] * TEMP_INV);
            const float e11 = __expf(c11[r] * TEMP_INV);
            acc0[r] += e00 + e01;
            acc1[r] += e10 + e11;
            if (j0 == i0) diag[i0] = e00;
            if (j1 == i0) diag[i0] = e01;
            if (j0 == i1) diag[i1] = e10;
            if (j1 == i1) diag[i1] = e11;
            if (j0 == p0) pos[i0] = e00;
            if (j1 == p0) pos[i0] = e01;
            if (j0 == p1) pos[i1] = e10;
            if (j1 == p1) pos[i1] = e11;
        }
#if HAVE_TDM
        __builtin_amdgcn_s_wait_tensorcnt(0);  // next buffer resident
#endif
        __syncthreads();                       // all waves done with Bs[cur]
    }

    // Reduce partial row sums across the 16 lanes of each half-wave, combine
    // the two col-group waves per row group via LDS atomics, then one global
    // atomic add per row (CSPLIT blocks contribute per row).
#pragma unroll
    for (int r = 0; r < 8; ++r) {
        float s0 = acc0[r], s1 = acc1[r];
#pragma unroll
        for (int m = 8; m >= 1; m >>= 1) {
            s0 += __shfl_xor(s0, m, 32);
            s1 += __shfl_xor(s1, m, 32);
        }
        if (lid == 0) {
            atomicAdd(&blockRowSum[wr * 32 + r + 8 * half], s0);
            atomicAdd(&blockRowSum[wr * 32 + 16 + r + 8 * half], s1);
        }
    }
    __syncthreads();
    if (tid < ROWS_PB) atomicAdd(&rowsum[rowBase + tid], blockRowSum[tid]);
}

// ---------------------------------------------------------------------------
// Kernel 3: loss = mean_i of -log(1e-5 + pos_i / (rowsum_i - diag_i))
// ---------------------------------------------------------------------------
__global__ void __launch_bounds__(256)
k_finalize(const float* __restrict__ rowsum,
           const float* __restrict__ diag,
           const float* __restrict__ pos,
           float* __restrict__ out) {
    __shared__ float red[8];
    const int tid = threadIdx.x;

    float local = 0.0f;
    for (int i = tid; i < N2; i += 256) {
        const float denom = rowsum[i] - diag[i];
        local += -__logf(1e-5f + pos[i] / denom);
    }
#pragma unroll
    for (int m = 16; m >= 1; m >>= 1) local += __shfl_xor(local, m, 32);
    if ((tid & 31) == 0) red[tid >> 5] = local;
    __syncthreads();
    if (tid == 0) {
        float t = 0.0f;
#pragma unroll
        for (int w = 0; w < 8; ++w) t += red[w];
        out[0] = t * (1.0f / (float)N2);
    }
}

// ---------------------------------------------------------------------------
// Launch. Workspace: rn (8 MB) | rowsum (32 KB) | diag (32 KB) | pos (32 KB).
// rowsum is zeroed by k_normalize (needed: CSPLIT blocks atomically add).
// diag/pos elements are each written exactly once per call.
// ---------------------------------------------------------------------------
extern "C" void kernel_launch(void* const* d_in, const int* in_sizes, int n_in,
                              void* d_out, int out_size, void* d_ws, size_t ws_size,
                              hipStream_t stream) {
    (void)in_sizes; (void)n_in; (void)out_size; (void)ws_size;

    const float* zis = (const float*)d_in[0];
    const float* zjs = (const float*)d_in[1];

    float* rn     = (float*)d_ws;
    float* rowsum = (float*)((char*)d_ws + (size_t)N2 * DDIM * sizeof(float));
    float* diag   = rowsum + N2;
    float* pos    = diag + N2;
    float* out    = (float*)d_out;

    k_normalize<<<N2 / 8, 256, 0, stream>>>(zis, zjs, rn, rowsum);
    k_gram<<<dim3(N2 / ROWS_PB, CSPLIT), 256, 0, stream>>>(rn, rowsum, diag, pos);
    k_finalize<<<1, 256, 0, stream>>>(rowsum, diag, pos, out);
}